// RelationalGraphConvolution_59339268161542
// MI455X (gfx1250) — compile-verified
//
#include <hip/hip_runtime.h>

#define N_NODES 100000
#define N_EDGES 1600000
#define D 128

typedef __attribute__((ext_vector_type(2))) float v2f;
typedef __attribute__((ext_vector_type(8))) float v8f;

// ---------------------------------------------------------------------------
// Zero an int array
// ---------------------------------------------------------------------------
__global__ void zero_int_kernel(int* __restrict__ p, int n) {
    int i = blockIdx.x * blockDim.x + threadIdx.x;
    if (i < n) p[i] = 0;
}

// ---------------------------------------------------------------------------
// Dual dense GEMM  h1 = x @ W1, h2 = x @ W2  via V_WMMA_F32_16X16X4_F32.
// One wave computes a 16x128 tile of BOTH outputs (shared A fragment -> x read once).
// Verified codegen: clauses of global_load_b32 overlapped with v_wmma under
// staggered s_wait_loadcnt.
// ---------------------------------------------------------------------------
__global__ __launch_bounds__(256) void gemm_dual_kernel(
        const float* __restrict__ x,
        const float* __restrict__ W1,
        const float* __restrict__ W2,
        float* __restrict__ h1,
        float* __restrict__ h2) {
    const int lane = threadIdx.x & 31;
    const int wid  = threadIdx.x >> 5;
    const int tile = blockIdx.x * 8 + wid;          // 16-row tile index
    const int m0   = tile * 16;
    if (m0 >= N_NODES) return;                      // wave-uniform exit (EXEC all-1 for WMMA)

    const int half = lane >> 4;                     // 0 or 1
    const int l15  = lane & 15;
    const int koff = half * 2;                      // per-lane K offset inside K=4 slab
    const float* arow = x + (size_t)(m0 + l15) * D;

    v8f acc1[8] = {};
    v8f acc2[8] = {};

    for (int k0 = 0; k0 < D; k0 += 4) {
        v2f a = *(const v2f*)(arow + k0 + koff);    // x[m0+l15, k0+koff .. +1]
        #pragma unroll
        for (int t = 0; t < 8; ++t) {
            const int n0 = t * 16;
            v2f b1, b2;
            b1.x = W1[(size_t)(k0 + koff    ) * D + n0 + l15];
            b1.y = W1[(size_t)(k0 + koff + 1) * D + n0 + l15];
            b2.x = W2[(size_t)(k0 + koff    ) * D + n0 + l15];
            b2.y = W2[(size_t)(k0 + koff + 1) * D + n0 + l15];
            acc1[t] = __builtin_amdgcn_wmma_f32_16x16x4_f32(
                false, a, false, b1, (short)0, acc1[t], false, false);
            acc2[t] = __builtin_amdgcn_wmma_f32_16x16x4_f32(
                false, a, false, b2, (short)0, acc2[t], false, false);
        }
    }

    #pragma unroll
    for (int t = 0; t < 8; ++t) {
        const int n0 = t * 16;
        #pragma unroll
        for (int r = 0; r < 8; ++r) {
            const int row = m0 + r + half * 8;
            h1[(size_t)row * D + n0 + l15] = acc1[t][r];
            h2[(size_t)row * D + n0 + l15] = acc2[t][r];
        }
    }
}

// ---------------------------------------------------------------------------
// CSR pass 1: degree histogram (native global_atomic_add_u32)
// ---------------------------------------------------------------------------
__global__ void hist_kernel(const int* __restrict__ dst, int* __restrict__ deg, int n) {
    int i = blockIdx.x * blockDim.x + threadIdx.x;
    if (i < n)
        __hip_atomic_fetch_add(&deg[dst[i]], 1, __ATOMIC_RELAXED, __HIP_MEMORY_SCOPE_AGENT);
}

// ---------------------------------------------------------------------------
// CSR pass 2: single-workgroup exclusive scan of degrees -> row offsets.
// Writes off[0..n] and initializes cursor[0..n-1] with the same offsets.
// deg may alias cursor (each thread only reads/writes its own index).
// ---------------------------------------------------------------------------
__global__ __launch_bounds__(1024) void exscan_kernel(
        const int* __restrict__ deg, int* __restrict__ off,
        int* __restrict__ cursor, int n) {
    __shared__ int smem[1024];
    __shared__ int carry;
    if (threadIdx.x == 0) carry = 0;
    __syncthreads();

    for (int base = 0; base < n; base += 1024) {
        const int i = base + (int)threadIdx.x;
        const int v = (i < n) ? deg[i] : 0;
        smem[threadIdx.x] = v;
        __syncthreads();
        for (int ofs = 1; ofs < 1024; ofs <<= 1) {
            int t = (threadIdx.x >= (unsigned)ofs) ? smem[threadIdx.x - ofs] : 0;
            __syncthreads();
            smem[threadIdx.x] += t;
            __syncthreads();
        }
        const int incl  = smem[threadIdx.x];
        const int total = smem[1023];
        const int excl  = carry + incl - v;
        if (i < n) { off[i] = excl; cursor[i] = excl; }
        __syncthreads();                 // all carry reads done
        if (threadIdx.x == 0) carry += total;
        __syncthreads();                 // carry update visible; smem reusable
    }
    if (threadIdx.x == 0) off[n] = carry;
}

// ---------------------------------------------------------------------------
// CSR pass 3: scatter packed (src, val) records into per-destination buckets.
// The gather pass then needs ONE sequential b64 load per edge instead of a
// random perm->src/vals double indirection.
// ---------------------------------------------------------------------------
__global__ void fill_csr_kernel(const int* __restrict__ dst, const int* __restrict__ src,
                                const float* __restrict__ vals, int* __restrict__ cursor,
                                int2* __restrict__ csr, int n) {
    int i = blockIdx.x * blockDim.x + threadIdx.x;
    if (i < n) {
        int slot = __hip_atomic_fetch_add(&cursor[dst[i]], 1,
                                          __ATOMIC_RELAXED, __HIP_MEMORY_SCOPE_AGENT);
        int2 rec;
        rec.x = src[i];
        rec.y = __float_as_int(vals[i]);
        csr[slot] = rec;
    }
}

// ---------------------------------------------------------------------------
// Pass 4: gather-SpMM for BOTH relations + fused relu, wave per node.
// Per edge: one wave-uniform b64 record load (sequential in CSR order) then
// the 512B h-row gather. Lane covers 4 columns. No fp atomics; out written
// exactly once, coalesced. h1/h2/out are all L2-resident (51 MB vs 192 MB L2).
// ---------------------------------------------------------------------------
__global__ __launch_bounds__(256) void gather_relu_kernel(
        const float* __restrict__ h1, const int* __restrict__ off1,
        const int2* __restrict__ csr1,
        const float* __restrict__ h2, const int* __restrict__ off2,
        const int2* __restrict__ csr2,
        float* __restrict__ out) {
    const int lane = threadIdx.x & 31;
    const int wid  = threadIdx.x >> 5;
    const int node = blockIdx.x * 8 + wid;
    if (node >= N_NODES) return;

    float4 acc = make_float4(0.0f, 0.0f, 0.0f, 0.0f);

    // relation 1
    {
        const int jb = off1[node], je = off1[node + 1];
        for (int j = jb; j < je; ++j) {
            const int2  rec = csr1[j];                             // wave-uniform
            const int   s   = __builtin_amdgcn_readfirstlane(rec.x);
            const float v   = __int_as_float(rec.y);
            const float4 hv = ((const float4*)(h1 + (size_t)s * D))[lane];
            acc.x = fmaf(v, hv.x, acc.x);
            acc.y = fmaf(v, hv.y, acc.y);
            acc.z = fmaf(v, hv.z, acc.z);
            acc.w = fmaf(v, hv.w, acc.w);
        }
    }
    // relation 2
    {
        const int jb = off2[node], je = off2[node + 1];
        for (int j = jb; j < je; ++j) {
            const int2  rec = csr2[j];
            const int   s   = __builtin_amdgcn_readfirstlane(rec.x);
            const float v   = __int_as_float(rec.y);
            const float4 hv = ((const float4*)(h2 + (size_t)s * D))[lane];
            acc.x = fmaf(v, hv.x, acc.x);
            acc.y = fmaf(v, hv.y, acc.y);
            acc.z = fmaf(v, hv.z, acc.z);
            acc.w = fmaf(v, hv.w, acc.w);
        }
    }

    acc.x = fmaxf(acc.x, 0.0f);
    acc.y = fmaxf(acc.y, 0.0f);
    acc.z = fmaxf(acc.z, 0.0f);
    acc.w = fmaxf(acc.w, 0.0f);
    ((float4*)(out + (size_t)node * D))[lane] = acc;
}

extern "C" void kernel_launch(void* const* d_in, const int* in_sizes, int n_in,
                              void* d_out, int out_size, void* d_ws, size_t ws_size,
                              hipStream_t stream) {
    const float* x     = (const float*)d_in[0];
    const float* W1    = (const float*)d_in[1];
    const float* W2    = (const float*)d_in[2];
    const float* vals1 = (const float*)d_in[3];
    const float* vals2 = (const float*)d_in[4];
    const int*   src1  = (const int*)d_in[5];
    const int*   dst1  = (const int*)d_in[6];
    const int*   src2  = (const int*)d_in[7];
    const int*   dst2  = (const int*)d_in[8];
    float* out = (float*)d_out;

    // workspace layout (~130 MB)
    float* h1   = (float*)d_ws;                       // 51.2 MB
    float* h2   = h1 + (size_t)N_NODES * D;           // 51.2 MB
    int2*  csr1 = (int2*)(h2 + (size_t)N_NODES * D);  // 12.8 MB packed (src,val)
    int2*  csr2 = csr1 + N_EDGES;                     // 12.8 MB
    int*   off1 = (int*)(csr2 + N_EDGES);             // 100001 ints
    int*   off2 = off1 + (N_NODES + 1);               // 100001 ints
    int*   cur1 = off2 + (N_NODES + 1);               // 100000 ints (also degree array)
    int*   cur2 = cur1 + N_NODES;                     // 100000 ints

    const int edgeBlocks = (N_EDGES + 255) / 256;     // 6250
    const int nodeBlocks = (N_NODES + 255) / 256;     // 391

    // dense projections (fp32 WMMA), independent of CSR build
    const int nTiles = N_NODES / 16;                  // 6250
    gemm_dual_kernel<<<(nTiles + 7) / 8, 256, 0, stream>>>(x, W1, W2, h1, h2);

    // build packed CSR for relation 1
    zero_int_kernel<<<nodeBlocks, 256, 0, stream>>>(cur1, N_NODES);
    hist_kernel<<<edgeBlocks, 256, 0, stream>>>(dst1, cur1, N_EDGES);
    exscan_kernel<<<1, 1024, 0, stream>>>(cur1, off1, cur1, N_NODES);
    fill_csr_kernel<<<edgeBlocks, 256, 0, stream>>>(dst1, src1, vals1, cur1, csr1, N_EDGES);

    // build packed CSR for relation 2
    zero_int_kernel<<<nodeBlocks, 256, 0, stream>>>(cur2, N_NODES);
    hist_kernel<<<edgeBlocks, 256, 0, stream>>>(dst2, cur2, N_EDGES);
    exscan_kernel<<<1, 1024, 0, stream>>>(cur2, off2, cur2, N_NODES);
    fill_csr_kernel<<<edgeBlocks, 256, 0, stream>>>(dst2, src2, vals2, cur2, csr2, N_EDGES);

    // fused gather-SpMM over both relations + relu (wave per node, no fp atomics)
    gather_relu_kernel<<<(N_NODES + 7) / 8, 256, 0, stream>>>(
        h1, off1, csr1, h2, off2, csr2, out);
}